// PhyGate_10746008175562
// MI455X (gfx1250) — compile-verified
//
#include <hip/hip_runtime.h>
#include <math.h>

// PhyGate: pure streaming elementwise op, B = 2,000,000 rows.
// 52 B/row of HBM traffic, ~75 flops/row -> bandwidth bound (~4.5 us floor
// at 23.3 TB/s). Strategy: 4 rows/thread so all global traffic is aligned
// b128, non-temporal (streamed once, keep L2 clean), scalars via SMEM.

typedef float f4 __attribute__((ext_vector_type(4)));

// One row of the reference math, f32 exactly as JAX computes it.
__device__ __forceinline__ void phygate_row(
    float bb, float vx, float vy, float vz,
    float wx, float wy, float wz,
    float lw, float lb, float dt,
    float p1, float p2, float p3x, float p3y, float p3z,
    float k /* p4*(1+p5) */, float p5, float r,
    float& vnx, float& vny, float& vnz,
    float& w2x, float& w2y, float& w2z)
{
    // gate1 = sigmoid(b*lin_w + lin_b)
    const float g1 = 1.0f / (1.0f + expf(-(bb * lw + lb)));
    const float g2 = 1.0f - g1;

    // angle-limited boundary term
    const float t1 = vx - wy * r;
    const float t2 = vy + wx * r;
    const float den = sqrtf(t1 * t1 + t2 * t2 + 1e-6f);
    float al = k * fabsf(vz) / den;
    al = (al > 0.4f) ? 0.4f : al;

    const float oma = 1.0f - al;
    const float vbx = oma * vx + al * r * wy;
    const float vby = oma * vy - al * r * wx;
    const float vbz = -p5 * vz;

    const float c    = 1.5f * al / r;
    const float om15 = 1.0f - 1.5f * al;
    const float wbx = -c * vy + om15 * wx;
    const float wby =  c * vx + om15 * wy;
    const float wbz =  wz;

    // gated blend
    const float v2x = g1 * vx + g2 * vbx;
    const float v2y = g1 * vy + g2 * vby;
    const float v2z = g1 * vz + g2 * vbz;
    w2x = g1 * wx + g2 * wbx;
    w2y = g1 * wy + g2 * wby;
    w2z = g1 * wz + g2 * wbz;

    // drag + Magnus + const accel, Euler step
    const float nv = sqrtf(v2x * v2x + v2y * v2y + v2z * v2z);
    const float cx = w2y * v2z - w2z * v2y;   // cross(w2, v2)
    const float cy = w2z * v2x - w2x * v2z;
    const float cz = w2x * v2y - w2y * v2x;
    const float ax = -p1 * nv * v2x + p2 * cx + p3x;
    const float ay = -p1 * nv * v2y + p2 * cy + p3y;
    const float az = -p1 * nv * v2z + p2 * cz + p3z;   // p3z includes -9.81

    vnx = v2x + ax * dt;
    vny = v2y + ay * dt;
    vnz = v2z + az * dt;
}

__global__ __launch_bounds__(256) void phygate_kernel(
    const float* __restrict__ b,  const float* __restrict__ v,  const float* __restrict__ w,
    const float* __restrict__ dtp, const float* __restrict__ lwp, const float* __restrict__ lbp,
    const float* __restrict__ p1p, const float* __restrict__ p2p, const float* __restrict__ p3p,
    const float* __restrict__ p4p, const float* __restrict__ p5p, const float* __restrict__ p6p,
    float* __restrict__ outv, float* __restrict__ outw, int n)
{
    const int tid = blockIdx.x * blockDim.x + threadIdx.x;
    const int n4  = n >> 2;

    // Uniform scalar params -> s_load (KMcnt), broadcast to all lanes.
    const float lw  = lwp[0], lb = lbp[0], dt = dtp[0];
    const float p1  = p1p[0], p2 = p2p[0];
    const float p3x = p3p[0], p3y = p3p[1], p3z = p3p[2] - 9.81f;
    const float p5  = p5p[0], r  = p6p[0];
    const float k   = p4p[0] * (1.0f + p5);

    if (tid < n4) {
        // ---- 4 rows per thread: all traffic is aligned b128, non-temporal.
        const f4 bq = __builtin_nontemporal_load((const f4*)b + tid);
        const f4* vp = (const f4*)v + 3 * tid;
        const f4* wp = (const f4*)w + 3 * tid;
        const f4 v0 = __builtin_nontemporal_load(vp + 0);
        const f4 v1 = __builtin_nontemporal_load(vp + 1);
        const f4 v2 = __builtin_nontemporal_load(vp + 2);
        const f4 w0 = __builtin_nontemporal_load(wp + 0);
        const f4 w1 = __builtin_nontemporal_load(wp + 1);
        const f4 w2 = __builtin_nontemporal_load(wp + 2);

        float vv[12], ww[12], ov[12], ow[12];
        #pragma unroll
        for (int j = 0; j < 4; ++j) {
            vv[j] = v0[j]; vv[4 + j] = v1[j]; vv[8 + j] = v2[j];
            ww[j] = w0[j]; ww[4 + j] = w1[j]; ww[8 + j] = w2[j];
        }

        #pragma unroll
        for (int i = 0; i < 4; ++i) {
            phygate_row(bq[i],
                        vv[3 * i], vv[3 * i + 1], vv[3 * i + 2],
                        ww[3 * i], ww[3 * i + 1], ww[3 * i + 2],
                        lw, lb, dt, p1, p2, p3x, p3y, p3z, k, p5, r,
                        ov[3 * i], ov[3 * i + 1], ov[3 * i + 2],
                        ow[3 * i], ow[3 * i + 1], ow[3 * i + 2]);
        }

        f4 a0, a1, a2, b0, b1, b2;
        #pragma unroll
        for (int j = 0; j < 4; ++j) {
            a0[j] = ov[j]; a1[j] = ov[4 + j]; a2[j] = ov[8 + j];
            b0[j] = ow[j]; b1[j] = ow[4 + j]; b2[j] = ow[8 + j];
        }
        f4* op = (f4*)outv + 3 * tid;
        f4* oq = (f4*)outw + 3 * tid;
        __builtin_nontemporal_store(a0, op + 0);
        __builtin_nontemporal_store(a1, op + 1);
        __builtin_nontemporal_store(a2, op + 2);
        __builtin_nontemporal_store(b0, oq + 0);
        __builtin_nontemporal_store(b1, oq + 1);
        __builtin_nontemporal_store(b2, oq + 2);
    } else if (tid == n4 && (n & 3)) {
        // ---- scalar tail (B = 2,000,000 is divisible by 4; kept for safety)
        for (int i = n4 * 4; i < n; ++i) {
            float vn0, vn1, vn2, wo0, wo1, wo2;
            phygate_row(b[i],
                        v[3 * i], v[3 * i + 1], v[3 * i + 2],
                        w[3 * i], w[3 * i + 1], w[3 * i + 2],
                        lw, lb, dt, p1, p2, p3x, p3y, p3z, k, p5, r,
                        vn0, vn1, vn2, wo0, wo1, wo2);
            outv[3 * i] = vn0; outv[3 * i + 1] = vn1; outv[3 * i + 2] = vn2;
            outw[3 * i] = wo0; outw[3 * i + 1] = wo1; outw[3 * i + 2] = wo2;
        }
    }
}

extern "C" void kernel_launch(void* const* d_in, const int* in_sizes, int n_in,
                              void* d_out, int out_size, void* d_ws, size_t ws_size,
                              hipStream_t stream)
{
    // setup_inputs() order: b, v, w, dt, lin_w, lin_b, param1..param6
    const float* b   = (const float*)d_in[0];
    const float* v   = (const float*)d_in[1];
    const float* w   = (const float*)d_in[2];
    const float* dt  = (const float*)d_in[3];
    const float* lw  = (const float*)d_in[4];
    const float* lb  = (const float*)d_in[5];
    const float* p1  = (const float*)d_in[6];
    const float* p2  = (const float*)d_in[7];
    const float* p3  = (const float*)d_in[8];
    const float* p4  = (const float*)d_in[9];
    const float* p5  = (const float*)d_in[10];
    const float* p6  = (const float*)d_in[11];

    const int n = in_sizes[0];          // B rows (b is (B,1))
    float* outv = (float*)d_out;                          // v_new: first 3*B
    float* outw = (float*)d_out + (size_t)3 * (size_t)n;  // w2:    next 3*B

    const int n4      = n >> 2;
    const int threads = n4 + ((n & 3) ? 1 : 0);
    const int block   = 256;
    const int grid    = (threads + block - 1) / block;

    phygate_kernel<<<grid, block, 0, stream>>>(b, v, w, dt, lw, lb,
                                               p1, p2, p3, p4, p5, p6,
                                               outv, outw, n);
}